// RSNN_TRSF_70205535420892
// MI455X (gfx1250) — compile-verified
//
#include <hip/hip_runtime.h>
#include <math.h>

// ---------------------------------------------------------------------------
// Problem constants (static per reference)
// ---------------------------------------------------------------------------
#define G_     16
#define W_     16
#define L_     64
#define NW     (G_*W_)        // 256 walks
#define NTOK   (NW*L_)        // 16384 tokens
#define D_     128            // d_model
#define HID    112
#define PE_IN  20
#define PE_OUT 16
#define FF     2048
#define FFC    512            // FFN column chunk (4 chunks)
#define NPER   500
#define TOTAL  (G_*NPER)      // 8000 static node segments (+1 trash)
#define EPS_   1e-5f

typedef __attribute__((ext_vector_type(2))) float v2f;
typedef __attribute__((ext_vector_type(8))) float v8f;

// ---------------------------------------------------------------------------
// Generic zero-fill
// ---------------------------------------------------------------------------
__global__ void k_zero(float* __restrict__ p, int n) {
  int i = blockIdx.x * blockDim.x + threadIdx.x;
  if (i < n) p[i] = 0.f;
}

// ---------------------------------------------------------------------------
// Embedding gather + PE projection + sinusoidal positional encoding
// ---------------------------------------------------------------------------
__global__ void k_embed(const int* __restrict__ walk_emb,
                        const float* __restrict__ walk_pe,
                        const float* __restrict__ emb,
                        const float* __restrict__ pe_w,
                        const float* __restrict__ pe_b,
                        float* __restrict__ x) {
  int t = blockIdx.x;          // token
  int d = threadIdx.x;         // feature
  int l = t & (L_ - 1);        // position within walk
  float v;
  if (d < HID) {
    int tok = walk_emb[t];
    v = emb[tok * HID + d];
  } else {
    int o = d - HID;
    float s = pe_b[o];
    const float* wp = walk_pe + (size_t)t * PE_IN;
    const float* ww = pe_w + o * PE_IN;
#pragma unroll
    for (int i = 0; i < PE_IN; ++i) s += wp[i] * ww[i];
    v = s;
  }
  int j = d >> 1;
  float ang = (float)l * expf(-(float)(2 * j) * (9.210340371976184f / 128.f));
  v += (d & 1) ? cosf(ang) : sinf(ang);
  x[(size_t)t * D_ + d] = v;
}

// ---------------------------------------------------------------------------
// FP32 WMMA GEMM:  C[M,N] (+)= A[M,K] @ B[N,K]^T  (+bias) (relu)
// Register-blocked: one wave owns a 32x64 C block = 2 M-tiles x 4 N-tiles,
// 8 x v8f accumulators; per K-step 2 A-frags + 4 B-frags feed 8
// V_WMMA_F32_16X16X4_F32 ops (10.7 FLOP/byte vs 4 for unblocked).
// Per ISA 7.12.2 32-bit A 16x4 layout: m=lane&15, khalf=(lane>>4)*2,
// a = {A[m][k+khalf], A[m][k+khalf+1]}; B (4x16) mirrored with n=lane&15.
// C/D: VGPR r holds row r (lanes 0-15) / r+8 (lanes 16-31), col = lane&15.
// flags: bit0 = relu, bit1 = accumulate into existing C.
// Requires M%32==0, N%64==0, K%4==0 (all call sites satisfy this).
// ---------------------------------------------------------------------------
#define WMMA4(a, b, c)                                                  \
  __builtin_amdgcn_wmma_f32_16x16x4_f32(false, (a), false, (b),         \
                                        (short)0, (c), false, false)

__global__ void k_gemm_wmma(const float* __restrict__ A, int lda,
                            const float* __restrict__ B, int ldb,
                            float* __restrict__ C, int ldc,
                            const float* __restrict__ bias,
                            int M, int N, int K, int flags) {
  const int lane = threadIdx.x & 31;
  const int wave = threadIdx.x >> 5;
  const int tN = N >> 6;                       // 64-col groups
  const int groups = (M >> 5) * tN;            // 32-row x 64-col blocks
  int gw = blockIdx.x * (blockDim.x >> 5) + wave;
  if (gw >= groups) return;                    // whole-wave exit: EXEC full
  int gm = gw / tN, gn = gw - gm * tN;

  const int l15    = lane & 15;
  const int khalf  = (lane >> 4) << 1;         // 0 or 2
  const int rowoff = (lane >> 4) << 3;         // 0 or 8
  const int mbase  = gm << 5;                  // 32 rows
  const int nbase  = gn << 6;                  // 64 cols

  const float* ap0 = A + (size_t)(mbase + l15) * lda + khalf;
  const float* ap1 = ap0 + (size_t)16 * lda;
  const float* bp0 = B + (size_t)(nbase + l15) * ldb + khalf;
  const size_t bstep = (size_t)16 * ldb;

  v8f acc[2][4];
#pragma unroll
  for (int i = 0; i < 2; ++i)
#pragma unroll
    for (int j = 0; j < 4; ++j)
      acc[i][j] = (v8f){0.f, 0.f, 0.f, 0.f, 0.f, 0.f, 0.f, 0.f};

  if (flags & 2) {
#pragma unroll
    for (int i = 0; i < 2; ++i)
#pragma unroll
      for (int j = 0; j < 4; ++j) {
        int ncol = nbase + (j << 4) + l15;
#pragma unroll
        for (int r = 0; r < 8; ++r)
          acc[i][j][r] =
              C[(size_t)(mbase + (i << 4) + r + rowoff) * ldc + ncol];
      }
  }

#pragma unroll 4
  for (int k = 0; k < K; k += 4) {
    v2f a0 = *(const v2f*)(ap0 + k);           // 8B-aligned pair loads
    v2f a1 = *(const v2f*)(ap1 + k);
    v2f b0 = *(const v2f*)(bp0 + k);
    v2f b1 = *(const v2f*)(bp0 + bstep + k);
    v2f b2 = *(const v2f*)(bp0 + 2 * bstep + k);
    v2f b3 = *(const v2f*)(bp0 + 3 * bstep + k);
    acc[0][0] = WMMA4(a0, b0, acc[0][0]);
    acc[0][1] = WMMA4(a0, b1, acc[0][1]);
    acc[0][2] = WMMA4(a0, b2, acc[0][2]);
    acc[0][3] = WMMA4(a0, b3, acc[0][3]);
    acc[1][0] = WMMA4(a1, b0, acc[1][0]);
    acc[1][1] = WMMA4(a1, b1, acc[1][1]);
    acc[1][2] = WMMA4(a1, b2, acc[1][2]);
    acc[1][3] = WMMA4(a1, b3, acc[1][3]);
  }

#pragma unroll
  for (int i = 0; i < 2; ++i)
#pragma unroll
    for (int j = 0; j < 4; ++j) {
      int ncol = nbase + (j << 4) + l15;
      float bb = bias ? bias[ncol] : 0.f;
#pragma unroll
      for (int r = 0; r < 8; ++r) {
        float v = acc[i][j][r] + bb;
        if (flags & 1) v = fmaxf(v, 0.f);
        C[(size_t)(mbase + (i << 4) + r + rowoff) * ldc + ncol] = v;
      }
    }
}

// ---------------------------------------------------------------------------
// head_dim==1 attention: for walk b, feature f:
//   attn[q,k] = softmax_k( q[b,q,f]*k[b,k,f] ) over valid keys (k < len[b])
//   o[b,q,f]  = sum_k attn * v[b,k,f]
// One block per (walk, 64-feature half); K/V half cached in LDS (32 KB).
// ---------------------------------------------------------------------------
__global__ void k_attn(const float* __restrict__ qkv,
                       const int* __restrict__ lengths,
                       float* __restrict__ o) {
  __shared__ float Ks[L_][64];
  __shared__ float Vs[L_][64];
  int b  = blockIdx.x >> 1;
  int fh = (blockIdx.x & 1) << 6;
  int tid = threadIdx.x;
  for (int idx = tid; idx < L_ * 64; idx += blockDim.x) {
    int kp = idx >> 6, f = idx & 63;
    size_t base = (size_t)(b * L_ + kp) * 384 + fh + f;
    Ks[kp][f] = qkv[base + 128];
    Vs[kp][f] = qkv[base + 256];
  }
  __syncthreads();
  int len = lengths[b];
  for (int idx = tid; idx < L_ * 64; idx += blockDim.x) {
    int qp = idx >> 6, f = idx & 63;
    float qv = qkv[(size_t)(b * L_ + qp) * 384 + fh + f];
    float mx = -3.4e38f;
    for (int k = 0; k < len; ++k) mx = fmaxf(mx, qv * Ks[k][f]);
    float den = 0.f, num = 0.f;
    for (int k = 0; k < len; ++k) {
      float e = expf(qv * Ks[k][f] - mx);
      den += e;
      num += e * Vs[k][f];
    }
    o[(size_t)(b * L_ + qp) * D_ + fh + f] = num / den;
  }
}

// ---------------------------------------------------------------------------
// x = LayerNorm(x + r) * w + b  (per-token, 128 features)
// ---------------------------------------------------------------------------
__global__ void k_add_ln(float* __restrict__ x, const float* __restrict__ r,
                         const float* __restrict__ w, const float* __restrict__ bb) {
  __shared__ float red[128];
  int t = blockIdx.x, d = threadIdx.x;
  float y = x[(size_t)t * D_ + d] + r[(size_t)t * D_ + d];
  red[d] = y;
  __syncthreads();
  for (int s = 64; s > 0; s >>= 1) {
    if (d < s) red[d] += red[d + s];
    __syncthreads();
  }
  float mean = red[0] * (1.f / 128.f);
  __syncthreads();
  float c = y - mean;
  red[d] = c * c;
  __syncthreads();
  for (int s = 64; s > 0; s >>= 1) {
    if (d < s) red[d] += red[d + s];
    __syncthreads();
  }
  float var = red[0] * (1.f / 128.f);
  x[(size_t)t * D_ + d] = c * rsqrtf(var + EPS_) * w[d] + bb[d];
}

// ---------------------------------------------------------------------------
// Per-graph max node id, then exclusive prefix-sum of (max+1) -> offsets
// ---------------------------------------------------------------------------
__global__ void k_gmax(const int* __restrict__ ids, int* __restrict__ gmax) {
  __shared__ int red[256];
  int g = blockIdx.x, tid = threadIdx.x;
  int m = (-0x7fffffff - 1);
  for (int i = tid; i < W_ * L_; i += 256) m = max(m, ids[g * W_ * L_ + i]);
  red[tid] = m;
  __syncthreads();
  for (int s = 128; s > 0; s >>= 1) {
    if (tid < s) red[tid] = max(red[tid], red[tid + s]);
    __syncthreads();
  }
  if (tid == 0) gmax[g] = red[0];
}

__global__ void k_offsets(const int* __restrict__ gmax, int* __restrict__ offs) {
  if (threadIdx.x == 0 && blockIdx.x == 0) {
    int acc = 0;
    for (int g = 0; g < G_; ++g) { offs[g] = acc; acc += gmax[g] + 1; }
  }
}

// ---------------------------------------------------------------------------
// Masked scatter-sum of token rows into node segments + counts
// ---------------------------------------------------------------------------
__global__ void k_scatter(const float* __restrict__ x, const int* __restrict__ ids,
                          const int* __restrict__ offs,
                          float* __restrict__ sums, float* __restrict__ cnt) {
  int t = blockIdx.x, d = threadIdx.x;
  int id = ids[t];
  if (id == -1) return;
  int g = t >> 10;                     // t / (W*L)
  int seg = id + offs[g];
  atomicAdd(&sums[(size_t)seg * D_ + d], x[(size_t)t * D_ + d]);
  if (d == 0) atomicAdd(&cnt[seg], 1.f);
}

__global__ void k_agg(const float* __restrict__ sums, const float* __restrict__ cnt,
                      float* __restrict__ agg) {
  int nd = blockIdx.x, d = threadIdx.x;
  agg[(size_t)nd * D_ + d] = sums[(size_t)nd * D_ + d] / fmaxf(cnt[nd], 1.f);
}

__global__ void k_scatter_back(float* __restrict__ x, const int* __restrict__ ids,
                               const int* __restrict__ offs,
                               const float* __restrict__ agg) {
  int t = blockIdx.x, d = threadIdx.x;
  int id = ids[t];
  if (id == -1) return;
  int g = t >> 10;
  int seg = id + offs[g];
  x[(size_t)t * D_ + d] = agg[(size_t)seg * D_ + d];
}

// ---------------------------------------------------------------------------
// Per-graph mean over NPER node rows
// ---------------------------------------------------------------------------
__global__ void k_gmean(const float* __restrict__ agg, float* __restrict__ gx) {
  int g = blockIdx.x, d = threadIdx.x;
  float s = 0.f;
  for (int i = 0; i < NPER; ++i) s += agg[(size_t)(g * NPER + i) * D_ + d];
  gx[g * D_ + d] = s * (1.f / (float)NPER);
}

// ---------------------------------------------------------------------------
// Readout head: h = relu(gx @ r1.T + b1); out = sigmoid(h @ r2.T + b2)
// ---------------------------------------------------------------------------
__global__ void k_head(const float* __restrict__ gx,
                       const float* __restrict__ r1w, const float* __restrict__ r1b,
                       const float* __restrict__ r2w, const float* __restrict__ r2b,
                       float* __restrict__ out) {
  __shared__ float h[G_ * D_];
  int tid = threadIdx.x;
  for (int idx = tid; idx < G_ * D_; idx += blockDim.x) {
    int g = idx >> 7, o = idx & 127;
    float s = r1b[o];
    for (int i = 0; i < D_; ++i) s += gx[g * D_ + i] * r1w[o * D_ + i];
    h[idx] = fmaxf(s, 0.f);
  }
  __syncthreads();
  for (int idx = tid; idx < G_ * 8; idx += blockDim.x) {
    int g = idx >> 3, o = idx & 7;
    float s = r2b[o];
    for (int i = 0; i < D_; ++i) s += h[g * D_ + i] * r2w[o * D_ + i];
    out[idx] = 1.f / (1.f + expf(-s));
  }
}

// ---------------------------------------------------------------------------
// Host orchestration
// Input order (recursive insertion order of setup_inputs()):
//  0 walk_emb  1 walk_ids  2 walk_pe  3 lengths  4 n_per_graph
//  5 emb  6 pe_w  7 pe_b
//  8+12*li : in_w,in_b,out_w,out_b,ln1_w,ln1_b,w1,b1,w2,b2,ln2_w,ln2_b
//  32 r1_w  33 r1_b  34 r2_w  35 r2_b
// ---------------------------------------------------------------------------
extern "C" void kernel_launch(void* const* d_in, const int* in_sizes, int n_in,
                              void* d_out, int out_size, void* d_ws, size_t ws_size,
                              hipStream_t stream) {
  (void)in_sizes; (void)n_in; (void)out_size; (void)ws_size;

  const int*   walk_emb = (const int*)d_in[0];
  const int*   walk_ids = (const int*)d_in[1];
  const float* walk_pe  = (const float*)d_in[2];
  const int*   lengths  = (const int*)d_in[3];
  const float* emb  = (const float*)d_in[5];
  const float* pe_w = (const float*)d_in[6];
  const float* pe_b = (const float*)d_in[7];
  const float* r1w  = (const float*)d_in[32];
  const float* r1b  = (const float*)d_in[33];
  const float* r2w  = (const float*)d_in[34];
  const float* r2b  = (const float*)d_in[35];

  // workspace bump allocator (256B aligned)
  char* wsp = (char*)d_ws;
  auto alloc = [&](size_t nbytes) -> void* {
    void* p = wsp;
    wsp += (nbytes + 255) & ~(size_t)255;
    return p;
  };
  float* x    = (float*)alloc((size_t)NTOK * D_ * 4);        //  8 MB
  float* qkv  = (float*)alloc((size_t)NTOK * 384 * 4);       // 24 MB
  float* ao   = (float*)alloc((size_t)NTOK * D_ * 4);        //  8 MB
  float* tmp  = (float*)alloc((size_t)NTOK * D_ * 4);        //  8 MB
  float* hch  = (float*)alloc((size_t)NTOK * FFC * 4);       // 32 MB
  float* sums = (float*)alloc((size_t)(TOTAL + 1) * D_ * 4); //  4 MB
  float* cnt  = (float*)alloc((size_t)(TOTAL + 1) * 4);
  float* agg  = (float*)alloc((size_t)TOTAL * D_ * 4);       //  4 MB
  int*   gmax = (int*)alloc(G_ * 4);
  int*   offs = (int*)alloc(G_ * 4);
  float* gx   = (float*)alloc(G_ * D_ * 4);

  auto gemm = [&](const float* A, int lda, const float* B, int ldb,
                  float* C, int ldc, const float* bias,
                  int M, int N, int K, int flags) {
    int groups = (M / 32) * (N / 64);           // 32x64 C block per wave
    int blocks = (groups + 7) / 8;              // 8 waves (256 threads) / block
    k_gemm_wmma<<<blocks, 256, 0, stream>>>(A, lda, B, ldb, C, ldc, bias,
                                            M, N, K, flags);
  };

  // node-id offsets (independent of x)
  k_gmax<<<G_, 256, 0, stream>>>(walk_ids, gmax);
  k_offsets<<<1, 32, 0, stream>>>(gmax, offs);

  // embedding + positional encodings
  k_embed<<<NTOK, 128, 0, stream>>>(walk_emb, walk_pe, emb, pe_w, pe_b, x);

  for (int li = 0; li < 2; ++li) {
    int base = 8 + li * 12;
    const float* in_w  = (const float*)d_in[base + 0];
    const float* in_b  = (const float*)d_in[base + 1];
    const float* out_w = (const float*)d_in[base + 2];
    const float* out_b = (const float*)d_in[base + 3];
    const float* ln1w  = (const float*)d_in[base + 4];
    const float* ln1b  = (const float*)d_in[base + 5];
    const float* w1    = (const float*)d_in[base + 6];
    const float* b1    = (const float*)d_in[base + 7];
    const float* w2    = (const float*)d_in[base + 8];
    const float* b2    = (const float*)d_in[base + 9];
    const float* ln2w  = (const float*)d_in[base + 10];
    const float* ln2b  = (const float*)d_in[base + 11];

    // qkv = x @ in_w.T + in_b      [16384 x 384]
    gemm(x, D_, in_w, D_, qkv, 384, in_b, NTOK, 384, D_, 0);
    // per-feature (head_dim=1) attention
    k_attn<<<NW * 2, 256, 0, stream>>>(qkv, lengths, ao);
    // out projection
    gemm(ao, D_, out_w, D_, tmp, D_, out_b, NTOK, D_, D_, 0);
    // x = LN(x + attn_out)
    k_add_ln<<<NTOK, 128, 0, stream>>>(x, tmp, ln1w, ln1b);

    // FFN in 4 column-chunks of 512: tmp = relu(x@w1.T+b1) @ w2.T + b2
    for (int c = 0; c < FF / FFC; ++c) {
      gemm(x, D_, w1 + (size_t)c * FFC * D_, D_, hch, FFC, b1 + c * FFC,
           NTOK, FFC, D_, /*relu*/1);
      gemm(hch, FFC, w2 + c * FFC, FF, tmp, D_,
           (c == 0) ? b2 : (const float*)nullptr,
           NTOK, D_, FFC, (c == 0) ? 0 : /*acc*/2);
    }
    // x = LN(x + ffn_out)
    k_add_ln<<<NTOK, 128, 0, stream>>>(x, tmp, ln2w, ln2b);

    // masked scatter-mean over global node ids
    k_zero<<<((TOTAL + 1) * D_ + 255) / 256, 256, 0, stream>>>(sums, (TOTAL + 1) * D_);
    k_zero<<<((TOTAL + 1) + 255) / 256, 256, 0, stream>>>(cnt, TOTAL + 1);
    k_scatter<<<NTOK, 128, 0, stream>>>(x, walk_ids, offs, sums, cnt);
    k_agg<<<TOTAL, 128, 0, stream>>>(sums, cnt, agg);
    if (li == 0)
      k_scatter_back<<<NTOK, 128, 0, stream>>>(x, walk_ids, offs, agg);
  }

  // per-graph mean + MLP head + sigmoid
  k_gmean<<<G_, 128, 0, stream>>>(agg, gx);
  k_head<<<1, 256, 0, stream>>>(gx, r1w, r1b, r2w, r2b, (float*)d_out);
}